// UnitaryBranching_35708358099360
// MI455X (gfx1250) — compile-verified
//
#include <hip/hip_runtime.h>

// ---------------------------------------------------------------------------
// UnitaryBranching on gfx1250 (MI455X).
// All 64x64 f32 matmuls run on V_WMMA_F32_16X16X4_F32 (wave32 WMMA).
// ---------------------------------------------------------------------------

typedef __attribute__((ext_vector_type(2))) float v2f;
typedef __attribute__((ext_vector_type(8))) float v8f;

#define DIM   64
#define NPOS  4096
#define NTHR  256   // 8 wave32s -> 16 tiles of 16x16, 2 tiles per wave

// C = A @ B for 64x64 f32 matrices. A, B read from LDS (row-major), C written
// to `Cout` (LDS or global, row-major). Must be called by all 256 threads with
// full EXEC (WMMA requirement).
__device__ __forceinline__ void mm64(const float* __restrict__ A,
                                     const float* __restrict__ B,
                                     float* __restrict__ Cout, int tid) {
  const int lane = tid & 31;
  const int wave = tid >> 5;        // wave32: 8 waves per block
  const int half = lane >> 4;       // 0: lanes 0-15, 1: lanes 16-31
  const int l16  = lane & 15;
#pragma unroll
  for (int tt = 0; tt < 2; ++tt) {
    const int t  = wave * 2 + tt;   // tile id 0..15
    const int mi = (t >> 2) * 16;   // tile row base
    const int ni = (t & 3) * 16;    // tile col base
    v8f acc = {};
#pragma unroll
    for (int k0 = 0; k0 < DIM; k0 += 4) {
      // A fragment 16x4: lanes 0-15 -> K = k0+{0,1}; lanes 16-31 -> K = k0+{2,3}
      const int ka = k0 + half * 2;
      v2f a;
      a.x = A[(mi + l16) * DIM + ka];
      a.y = A[(mi + l16) * DIM + ka + 1];
      // B fragment 4x16: VGPR r holds row K = ka + r for this lane half
      v2f b;
      b.x = B[ka * DIM + ni + l16];
      b.y = B[(ka + 1) * DIM + ni + l16];
      // 8-arg form: (neg_a, A, neg_b, B, c_mod, C, reuse_a, reuse_b)
      acc = __builtin_amdgcn_wmma_f32_16x16x4_f32(
          false, a, false, b, (short)0, acc, false, false);
    }
    // C layout: VGPR r -> row mi + r + half*8, col ni + l16
    const int col = ni + l16;
    const int rbase = mi + half * 8;
#pragma unroll
    for (int r = 0; r < 8; ++r) Cout[(rbase + r) * DIM + col] = acc[r];
  }
}

// ---------------------------------------------------------------------------
// Stage A: expm of the two used generators, store transposed into ws.
// expm via scaling-and-squaring (squarings=8) + Horner Taylor (order=12),
// exactly mirroring the reference.
// ---------------------------------------------------------------------------
__global__ void __launch_bounds__(NTHR)
expm_kernel(const float* __restrict__ raw, float* __restrict__ ext) {
  __shared__ float As[DIM * DIM];
  __shared__ float E[DIM * DIM];
  __shared__ float Tm[DIM * DIM];
  const int tid = threadIdx.x;
  const float* src = raw + (size_t)blockIdx.x * DIM * DIM;

  // As = (raw - raw^T) / 2^8 ; E = I
  for (int idx = tid; idx < DIM * DIM; idx += NTHR) {
    const int i = idx >> 6, j = idx & 63;
    As[idx] = (src[idx] - src[j * DIM + i]) * 0.00390625f;  // * 2^-8
    E[idx]  = (i == j) ? 1.0f : 0.0f;
  }
  __syncthreads();

  // Horner: for k = 12..1: E = I + (As @ E) / k
  for (int k = 12; k >= 1; --k) {
    mm64(As, E, Tm, tid);
    __syncthreads();
    for (int idx = tid; idx < DIM * DIM; idx += NTHR) {
      const int i = idx >> 6, j = idx & 63;
      E[idx] = ((i == j) ? 1.0f : 0.0f) + Tm[idx] / (float)k;
    }
    __syncthreads();
  }
  // 8 squarings: E = E @ E
  for (int s = 0; s < 8; ++s) {
    mm64(E, E, Tm, tid);
    __syncthreads();
    for (int idx = tid; idx < DIM * DIM; idx += NTHR) E[idx] = Tm[idx];
    __syncthreads();
  }
  // ext[h] = expm(H_h)^T
  float* dst = ext + (size_t)blockIdx.x * DIM * DIM;
  for (int idx = tid; idx < DIM * DIM; idx += NTHR) {
    const int i = idx >> 6, j = idx & 63;
    dst[idx] = E[j * DIM + i];
  }
}

// ---------------------------------------------------------------------------
// Stage B: binary-tree recurrence  M[node] = M[(node-1)/2] @ ext[(node-1)%2],
// one 64x64 WMMA matmul per block, launched level by level.
// ---------------------------------------------------------------------------
__global__ void __launch_bounds__(NTHR)
level_kernel(const float* __restrict__ maps_in, const float* __restrict__ ext,
             float* __restrict__ maps_out, int level_start) {
  __shared__ float A[DIM * DIM];
  __shared__ float B[DIM * DIM];
  const int node   = level_start + blockIdx.x;
  const int parent = (node - 1) >> 1;
  const int digit  = (node - 1) & 1;
  const int tid = threadIdx.x;
  const float* Pa = maps_in + (size_t)parent * DIM * DIM;
  const float* Ex = ext + (size_t)digit * DIM * DIM;
  for (int idx = tid; idx < DIM * DIM; idx += NTHR) {
    A[idx] = Pa[idx];
    B[idx] = Ex[idx];
  }
  __syncthreads();
  mm64(A, B, maps_out + (size_t)node * DIM * DIM, tid);
}

// ---------------------------------------------------------------------------
// Keys: pack each (padded) path word into 2-bit fields, MSB-first, so the
// common-prefix length is clz(xor)>>1. Also write maps[0] = I.
// ---------------------------------------------------------------------------
__global__ void __launch_bounds__(NTHR)
keys_kernel(const int* __restrict__ path_words, int T,
            unsigned* __restrict__ keys, float* __restrict__ maps0) {
  const int i = blockIdx.x * blockDim.x + threadIdx.x;
  if (i < NPOS) {
    unsigned key = 0u;
    const int tmax = (T < 15) ? T : 15;
    for (int t = 0; t < tmax; ++t)
      key |= ((unsigned)(path_words[(size_t)i * T + t] & 3)) << (30 - 2 * t);
    keys[i] = key;
  }
  if (blockIdx.x == 0) {
    for (int idx = threadIdx.x; idx < DIM * DIM; idx += NTHR)
      maps0[idx] = ((idx >> 6) == (idx & 63)) ? 1.0f : 0.0f;
  }
}

// ---------------------------------------------------------------------------
// Steps: steps[i,j] = 2T - 2*commonPrefix(word_i, word_j).
// (Reference's mask tests != bf+1 while padding is bf, so lens == T always.)
// ---------------------------------------------------------------------------
__global__ void __launch_bounds__(NTHR)
steps_kernel(const unsigned* __restrict__ keys, float* __restrict__ steps, int T) {
  const int i = blockIdx.x;
  const unsigned ki = keys[i];
  float* row = steps + (size_t)i * NPOS;
  for (int j = threadIdx.x; j < NPOS; j += blockDim.x) {
    const unsigned x = ki ^ keys[j];
    int common = (x == 0u) ? T : (__clz((int)x) >> 1);
    if (common > T) common = T;
    row[j] = (float)(2 * T - 2 * common);
  }
}

// ---------------------------------------------------------------------------
extern "C" void kernel_launch(void* const* d_in, const int* in_sizes, int n_in,
                              void* d_out, int out_size, void* d_ws, size_t ws_size,
                              hipStream_t stream) {
  const float* raw = (const float*)d_in[0];       // (17, 64, 64) f32
  const int*   pw  = (const int*)d_in[1];         // (4096, T) int32
  const int T = in_sizes[1] / NPOS;               // = 12

  float* maps  = (float*)d_out;                   // (4096, 1, 64, 64) f32
  float* steps = maps + (size_t)NPOS * DIM * DIM; // (4096, 4096) as f32

  float*    ext  = (float*)d_ws;                                   // 2*64*64 f32
  unsigned* keys = (unsigned*)((char*)d_ws + 2 * DIM * DIM * sizeof(float));

  // Stage A: two matrix exponentials (transposed) -> ws
  expm_kernel<<<2, NTHR, 0, stream>>>(raw, ext);

  // Keys + maps[0] = I
  keys_kernel<<<(NPOS + NTHR - 1) / NTHR, NTHR, 0, stream>>>(pw, T, keys, maps);

  // Stage B: tree levels (level l: nodes [2^l - 1, 2^(l+1) - 2])
  int level_start = 1, count = 2;
  while (level_start < NPOS) {
    int n = count;
    if (level_start + n > NPOS) n = NPOS - level_start;
    level_kernel<<<n, NTHR, 0, stream>>>(maps, ext, maps, level_start);
    level_start += count;
    count <<= 1;
  }

  // Stage C: pairwise step distances
  steps_kernel<<<NPOS, NTHR, 0, stream>>>(keys, steps, T);
}